// BasisSlotAffinityGAT_84258668413231
// MI455X (gfx1250) — compile-verified
//
#include <hip/hip_runtime.h>
#include <math.h>

// ---------------------------------------------------------------------------
// Problem constants (match reference)
// ---------------------------------------------------------------------------
#define Bb   8
#define Nn   512
#define Dd   256
#define Hh   8
#define Mm   8
#define Kk   16
#define Rr   8
#define DHh  32
#define BN_  4096       // B*N
#define BH_  64         // B*H
#define BHM_ 512        // B*H*M

#define LOGIT_SCALE 0.0625f     // (1/sqrt(D)) / TAU_AFF
#define A_W  (1.0f/512.0f)
#define B_W  (1.0f/16.0f)
#define GW_INV_EPS 20.0f        // 1/0.05
#define SINK_TOL 1e-6f

// d_out layout: bias_log [64*512*512], Q [64*512*512], total_reg [1]
#define OUT_Q   16777216u
#define OUT_REG 33554432u

// workspace layout (float offsets)
#define OF_XCAT 0u                 // 4096*512            = 2,097,152
#define OF_H1   2097152u           // 4096*256            = 1,048,576
#define OF_Z    3145728u           // 4096*256
#define OF_ZN   4194304u           // 4096*256  ([B,H,N,DH])
#define OF_P    5242880u           // 64*512*512          = 16,777,216
#define OF_DP   22020096u          // 64*512*512
// reuse inside P region (P dead after k_sym_dp):
#define OF_PI   OF_P               // 512*512*16 = 4,194,304
#define OF_C    9437184u
#define OF_T    13631488u          // E buffer during GW loop, final T after
// reuse inside XCAT region (dead after MLP):
#define OF_MJ   0u                 // [B,H,M,N] 262,144
#define OF_T1   262144u            // [B,H,M,N]
#define OF_ML   524288u            // [B,H,M,K] 8,192
#define OF_T2   532480u            // [B,H,M,K]
#define OF_GW   540672u            // [B,H,M]   512
#define OF_AL   541184u            // [B,H,M]
#define OF_US   541696u            // [M]
// reuse inside H1 region (dead after second GEMM):
#define OF_G    2097152u           // [M,K,K] 2048
#define OF_DG   2099200u
#define OF_DY2  2101248u
#define OF_GREG 2103296u           // scalar

typedef __attribute__((ext_vector_type(2))) float v2f;
typedef __attribute__((ext_vector_type(8))) float v8f;

// D(16x16,f32) = A(16x4,f32) * B(4x16,f32) + C  -- CDNA5 v_wmma_f32_16x16x4_f32
static __device__ __forceinline__ v8f wmma4(v2f a, v2f b, v8f c) {
  return __builtin_amdgcn_wmma_f32_16x16x4_f32(false, a, false, b, (short)0, c,
                                               false, false);
}

static __device__ __forceinline__ float wave_sum(float v) {
  for (int off = 16; off > 0; off >>= 1) v += __shfl_xor(v, off, 32);
  return v;
}
static __device__ __forceinline__ float wave_max(float v) {
  for (int off = 16; off > 0; off >>= 1) v = fmaxf(v, __shfl_xor(v, off, 32));
  return v;
}
// all 256 threads must call
static __device__ __forceinline__ float block_sum(float v, float* sm) {
  int lane = threadIdx.x & 31, wid = threadIdx.x >> 5;
  v = wave_sum(v);
  if (lane == 0) sm[wid] = v;
  __syncthreads();
  float r = (threadIdx.x < 8) ? sm[threadIdx.x] : 0.f;
  if (wid == 0) {
    r = wave_sum(r);
    if (lane == 0) sm[0] = r;
  }
  __syncthreads();
  r = sm[0];
  __syncthreads();
  return r;
}

// ---------------------------------------------------------------------------
// Stage A: fusion MLP
// ---------------------------------------------------------------------------
__global__ void k_concat(const float* __restrict__ xd, const float* __restrict__ xn,
                         float* __restrict__ xc) {
  int i = blockIdx.x * 256 + threadIdx.x;          // 4096*512
  int r = i >> 9, c = i & 511;
  xc[i] = (c < 256) ? xd[r * 256 + c] : xn[r * 256 + (c - 256)];
}

// C[4096,Ncols] = act(A[4096,Kdim] @ Bm[Kdim,Ncols] + bias); one wave per 16x16 tile
__global__ void k_gemm(const float* __restrict__ A, const float* __restrict__ Bm,
                       const float* __restrict__ bias, float* __restrict__ Co,
                       int Kdim, int Ncols, int relu) {
  int wid = threadIdx.x >> 5, lane = threadIdx.x & 31;
  int hf = lane >> 4, l16 = lane & 15;
  int tile = blockIdx.x * 8 + wid;
  int nt = Ncols >> 4;
  int r0 = (tile / nt) * 16, c0 = (tile % nt) * 16;
  v8f acc = {};
  for (int k0 = 0; k0 < Kdim; k0 += 4) {
    v2f a, b;
    a.x = A[(size_t)(r0 + l16) * Kdim + k0 + 2 * hf];
    a.y = A[(size_t)(r0 + l16) * Kdim + k0 + 2 * hf + 1];
    b.x = Bm[(size_t)(k0 + 2 * hf) * Ncols + c0 + l16];
    b.y = Bm[(size_t)(k0 + 2 * hf + 1) * Ncols + c0 + l16];
    acc = wmma4(a, b, acc);
  }
  float bv = bias[c0 + l16];
  for (int g = 0; g < 8; ++g) {
    float v = acc[g] + bv;
    if (relu) v = fmaxf(v, 0.f);
    Co[(size_t)(r0 + g + 8 * hf) * Ncols + c0 + l16] = v;
  }
}

// ---------------------------------------------------------------------------
// Stage B: per-head l2 normalize, affinity softmax, symmetrize -> DP
// ---------------------------------------------------------------------------
__global__ void k_znorm(const float* __restrict__ Z, float* __restrict__ ZN) {
  int wid = threadIdx.x >> 5, lane = threadIdx.x & 31;
  int w = blockIdx.x * 8 + wid;       // (b*8+h)*512+n, 0..32767
  int b = w >> 12, h = (w >> 9) & 7, n = w & 511;
  float v = Z[((size_t)(b * 512 + n)) * 256 + h * 32 + lane];
  float ss = wave_sum(v * v);
  float nrm = fmaxf(sqrtf(ss), 1e-8f);
  ZN[(size_t)w * 32 + lane] = v / nrm;
}

// block = (bh, 16-row chunk); waves tile 512 cols; then row softmax in LDS
__global__ void k_affinity(const float* __restrict__ ZN, float* __restrict__ P) {
  __shared__ float S[16 * 512];
  int bh = blockIdx.x >> 5, rchunk = blockIdx.x & 31;
  int wid = threadIdx.x >> 5, lane = threadIdx.x & 31;
  int hf = lane >> 4, l16 = lane & 15;
  const float* zb = ZN + (size_t)bh * Nn * DHh;
  int r0 = rchunk * 16;
  for (int q = 0; q < 4; ++q) {
    int c0 = (q * 8 + wid) * 16;
    v8f acc = {};
    for (int k0 = 0; k0 < DHh; k0 += 4) {
      v2f a, b;
      a.x = zb[(size_t)(r0 + l16) * DHh + k0 + 2 * hf];
      a.y = zb[(size_t)(r0 + l16) * DHh + k0 + 2 * hf + 1];
      b.x = zb[(size_t)(c0 + l16) * DHh + k0 + 2 * hf];
      b.y = zb[(size_t)(c0 + l16) * DHh + k0 + 2 * hf + 1];
      acc = wmma4(a, b, acc);
    }
    for (int g = 0; g < 8; ++g) {
      int i = g + 8 * hf;                 // local row
      int j = c0 + l16;                   // global col
      float v = acc[g] * LOGIT_SCALE;
      if (r0 + i == j) v = -1e9f;         // no self loop
      S[i * 512 + j] = v;
    }
  }
  __syncthreads();
  for (int rr = wid; rr < 16; rr += 8) {
    float mx = -3.4e38f;
    for (int c = lane; c < 512; c += 32) mx = fmaxf(mx, S[rr * 512 + c]);
    mx = wave_max(mx);
    float sum = 0.f;
    for (int c = lane; c < 512; c += 32) {
      float e = expf(S[rr * 512 + c] - mx);
      S[rr * 512 + c] = e;
      sum += e;
    }
    sum = wave_sum(sum);
    float inv = 1.f / sum;
    float* prow = P + (size_t)bh * Nn * Nn + (size_t)(r0 + rr) * Nn;
    for (int c = lane; c < 512; c += 32) prow[c] = S[rr * 512 + c] * inv;
  }
}

__global__ void k_sym_dp(const float* __restrict__ P, float* __restrict__ DPo) {
  __shared__ float red[8];
  int bh = blockIdx.x >> 5, rchunk = blockIdx.x & 31;
  const float* pb = P + (size_t)bh * Nn * Nn;
  float* db = DPo + (size_t)bh * Nn * Nn;
  int t = threadIdx.x;
  for (int r = 0; r < 16; ++r) {
    int i = rchunk * 16 + r;
    int c0 = t, c1 = t + 256;
    float v0 = fmaxf(0.5f * (pb[(size_t)i * 512 + c0] + pb[(size_t)c0 * 512 + i]), 1e-8f);
    float v1 = fmaxf(0.5f * (pb[(size_t)i * 512 + c1] + pb[(size_t)c1 * 512 + i]), 1e-8f);
    float s = block_sum(v0 + v1, red);
    float inv = 1.f / fmaxf(s, 1e-8f);
    db[(size_t)i * 512 + c0] = fmaxf(1.f - v0 * inv, 0.f);
    db[(size_t)i * 512 + c1] = fmaxf(1.f - v1 * inv, 0.f);
  }
}

// ---------------------------------------------------------------------------
// Stage C: slot graphs G, diversity reg, slot cost DG, Dy2 (one block)
// ---------------------------------------------------------------------------
__global__ void k_slots(const float* __restrict__ Up, float* __restrict__ G,
                        float* __restrict__ DG, float* __restrict__ Dy2,
                        float* __restrict__ greg) {
  __shared__ float Us[1024];
  __shared__ float Gs[2048];
  __shared__ float nrm[8];
  __shared__ float gram[64];
  int t = threadIdx.x;
  for (int i = t; i < 1024; i += 256) {
    float x = Up[i];
    Us[i] = (x > 20.f) ? x : log1pf(expf(x));        // softplus
  }
  __syncthreads();
  for (int i = t; i < 2048; i += 256) {
    int m = i >> 8, k = (i >> 4) & 15, j = i & 15;
    const float* uk = &Us[m * 128 + k * 8];
    const float* uj = &Us[m * 128 + j * 8];
    float s = 0.f;
    for (int r = 0; r < 8; ++r) s += uk[r] * uj[r];
    Gs[i] = s;
  }
  __syncthreads();
  if (t < 128) {                                     // row-stochastic
    float s = 0.f;
    for (int j = 0; j < 16; ++j) s += Gs[t * 16 + j];
    float inv = 1.f / fmaxf(s, 1e-8f);
    for (int j = 0; j < 16; ++j) Gs[t * 16 + j] *= inv;
  }
  __syncthreads();
  for (int i = t; i < 2048; i += 256) G[i] = Gs[i];
  if (t < 8) {
    float s = 0.f;
    for (int i = 0; i < 256; ++i) { float v = Gs[t * 256 + i]; s += v * v; }
    nrm[t] = fmaxf(sqrtf(s), 1e-8f);
  }
  __syncthreads();
  if (t < 64) {
    int p = t >> 3, q = t & 7;
    float s = 0.f;
    for (int i = 0; i < 256; ++i) s += Gs[p * 256 + i] * Gs[q * 256 + i];
    gram[t] = s / (nrm[p] * nrm[q]);
  }
  __syncthreads();
  if (t == 0) {
    float off = 0.f;
    for (int p = 0; p < 8; ++p)
      for (int q = 0; q < 8; ++q)
        if (p != q) off += gram[p * 8 + q];
    greg[0] = 0.1f * off / 56.f;                     // lambda * sum / (M*(M-1))
  }
  for (int i = t; i < 2048; i += 256) {
    int m = i >> 8, k = (i >> 4) & 15, j = i & 15;
    float dk = fmaxf(Gs[m * 256 + k * 16 + k], 1e-8f);
    float dj = fmaxf(Gs[m * 256 + j * 16 + j], 1e-8f);
    float den = fmaxf(sqrtf(dk * dj), 1e-8f);
    float c = fminf(fmaxf(Gs[i] / den, -1.f), 1.f);
    float d = fmaxf(1.f - c, 0.f);
    if (k == j) d = 0.f;
    DG[i] = d;
    Dy2[i] = d * d;
  }
}

// ---------------------------------------------------------------------------
// Stage D: entropic Gromov-Wasserstein loop
// ---------------------------------------------------------------------------
__global__ void k_fill(float* __restrict__ p, float val, int n) {
  int i = blockIdx.x * 256 + threadIdx.x;
  if (i < n) p[i] = val;
}

// block per (b,h,m): mass_j[n] = row sums, mass_l[k] = col sums of Pi
__global__ void k_mass(const float* __restrict__ Pi, float* __restrict__ mj,
                       float* __restrict__ ml) {
  __shared__ float cacc[16];
  int bhm = blockIdx.x, t = threadIdx.x;
  if (t < 16) cacc[t] = 0.f;
  __syncthreads();
  const float* pb = Pi + (size_t)bhm * Nn * Kk;
  float part[16];
  for (int k = 0; k < 16; ++k) part[k] = 0.f;
  for (int r = t; r < 512; r += 256) {
    float s = 0.f;
    for (int k = 0; k < 16; ++k) {
      float p = pb[r * 16 + k];
      s += p;
      part[k] += p;
    }
    mj[(size_t)bhm * Nn + r] = s;
  }
  for (int k = 0; k < 16; ++k) atomicAdd(&cacc[k], part[k]);
  __syncthreads();
  if (t < 16) ml[(size_t)bhm * Kk + t] = cacc[t];
}

__global__ void k_term2(const float* __restrict__ Dy2, const float* __restrict__ ml,
                        float* __restrict__ t2) {
  int idx = blockIdx.x * 256 + threadIdx.x;   // 8192 = BHM*K
  int bhm = idx >> 4, k = idx & 15, m = bhm & 7;
  const float* dy = Dy2 + m * 256 + k * 16;
  const float* mb = ml + (size_t)bhm * 16;
  float s = 0.f;
  for (int l = 0; l < 16; ++l) s += dy[l] * mb[l];
  t2[idx] = s;
}

// term1[b,h,m,i] = sum_j DP^2[i,j] * mass_j[m,j] ; block per (bh, 16-row chunk)
__global__ void k_term1(const float* __restrict__ DPg, const float* __restrict__ mjg,
                        float* __restrict__ t1) {
  __shared__ float dx2[16 * 512];
  __shared__ float mj[8 * 512];
  int bh = blockIdx.x >> 5, rchunk = blockIdx.x & 31;
  int t = threadIdx.x;
  const float* db = DPg + (size_t)bh * Nn * Nn + (size_t)rchunk * 16 * Nn;
  for (int i = t; i < 16 * 512; i += 256) { float v = db[i]; dx2[i] = v * v; }
  const float* mb = mjg + (size_t)bh * Mm * Nn;
  for (int i = t; i < 8 * 512; i += 256) mj[i] = mb[i];
  __syncthreads();
  if (t < 128) {
    int m = t >> 4, il = t & 15;
    float s = 0.f;
    for (int j = 0; j < 512; ++j) s += dx2[il * 512 + j] * mj[m * 512 + j];
    t1[((size_t)bh * Mm + m) * Nn + rchunk * 16 + il] = s;
  }
}

// E[b,h,m,n,k] = sum_l Pi[n,l] * DG[m,k,l]
__global__ void k_piDG(const float* __restrict__ Pi, const float* __restrict__ DGg,
                       float* __restrict__ E) {
  __shared__ float dgs[256];
  int bhm = blockIdx.x >> 1, hh = blockIdx.x & 1;
  int m = bhm & 7, t = threadIdx.x;
  dgs[t] = DGg[m * 256 + t];
  __syncthreads();
  int n = hh * 256 + t;
  const float* pr = Pi + ((size_t)bhm * Nn + n) * Kk;
  float row[16];
  for (int l = 0; l < 16; ++l) row[l] = pr[l];
  float* er = E + ((size_t)bhm * Nn + n) * Kk;
  for (int k = 0; k < 16; ++k) {
    float s = 0.f;
    for (int l = 0; l < 16; ++l) s += row[l] * dgs[k * 16 + l];
    er[k] = s;
  }
}

// cross = DP @ E; C = clip(term1 + term2 - 2*cross, 0)
// block per (bh, itile); 8 waves = 8 slot graphs m sharing LDS-staged DP tile
__global__ void k_cross(const float* __restrict__ DPg, const float* __restrict__ E,
                        const float* __restrict__ t1g, const float* __restrict__ t2g,
                        float* __restrict__ Cb) {
  __shared__ float at[16 * 68];
  int bh = blockIdx.x >> 5, itile = blockIdx.x & 31;
  int wid = threadIdx.x >> 5, lane = threadIdx.x & 31;
  int hf = lane >> 4, l16 = lane & 15;
  int i0 = itile * 16;
  size_t bhm = (size_t)bh * 8 + wid;
  const float* db = DPg + (size_t)bh * Nn * Nn;
  const float* eb = E + bhm * Nn * Kk;
  v8f acc = {};
  for (int k0 = 0; k0 < 512; k0 += 64) {
    __syncthreads();
    for (int i = threadIdx.x; i < 16 * 64; i += 256) {
      int r = i >> 6, c = i & 63;
      at[r * 68 + c] = db[(size_t)(i0 + r) * Nn + k0 + c];
    }
    __syncthreads();
    for (int kk = 0; kk < 16; ++kk) {
      int kl = kk * 4 + 2 * hf;
      v2f a, b;
      a.x = at[l16 * 68 + kl];
      a.y = at[l16 * 68 + kl + 1];
      const float* ep = eb + (size_t)(k0 + kl) * Kk;
      b.x = ep[l16];
      b.y = ep[Kk + l16];
      acc = wmma4(a, b, acc);
    }
  }
  const float* t1 = t1g + bhm * Nn;
  float t2v = t2g[bhm * Kk + l16];
  float* cb = Cb + bhm * Nn * Kk;
  for (int g = 0; g < 8; ++g) {
    int i = i0 + g + 8 * hf;
    cb[(size_t)i * Kk + l16] = fmaxf(t1[i] + t2v - 2.f * acc[g], 0.f);
  }
}

// per-(b,h,m) Sinkhorn, 10 iterations, entirely in LDS; writes Pi and gw
__global__ void k_sinkhorn(const float* __restrict__ Cb, float* __restrict__ Pi,
                           float* __restrict__ gw) {
  __shared__ float Km[512 * 16];
  __shared__ float u[512];
  __shared__ float v[16];
  __shared__ float vacc[16];
  __shared__ float red[8];
  int bhm = blockIdx.x, t = threadIdx.x;
  const float* cb = Cb + (size_t)bhm * Nn * Kk;
  for (int i = t; i < 512 * 16; i += 256)
    Km[i] = fmaxf(expf(-cb[i] * GW_INV_EPS), 1e-12f);
  u[t] = A_W;
  u[t + 256] = A_W;
  if (t < 16) v[t] = B_W;
  __syncthreads();
  for (int it = 0; it < 10; ++it) {
    for (int r = t; r < 512; r += 256) {
      float s = 0.f;
      for (int k = 0; k < 16; ++k) s += Km[r * 16 + k] * v[k];
      u[r] = A_W / (s + SINK_TOL);
    }
    __syncthreads();
    if (t < 16) vacc[t] = 0.f;
    __syncthreads();
    for (int r = t; r < 512; r += 256) {
      float ur = u[r];
      for (int k = 0; k < 16; ++k) atomicAdd(&vacc[k], Km[r * 16 + k] * ur);
    }
    __syncthreads();
    if (t < 16) v[t] = B_W / (vacc[t] + SINK_TOL);
    __syncthreads();
  }
  float lsum = 0.f;
  float* pb = Pi + (size_t)bhm * Nn * Kk;
  for (int r = t; r < 512; r += 256) {
    float ur = u[r];
    for (int k = 0; k < 16; ++k) {
      float p = ur * Km[r * 16 + k] * v[k];
      pb[r * 16 + k] = p;
      lsum += p * cb[r * 16 + k];
    }
  }
  float s = block_sum(lsum, red);
  if (t == 0) gw[bhm] = s;
}

// ---------------------------------------------------------------------------
// Stage E: alpha, T = Pi@G, Qmix via WMMA, normalize -> Q and bias_log
// ---------------------------------------------------------------------------
__global__ void k_alpha(const float* __restrict__ gw, float* __restrict__ al) {
  int bh = blockIdx.x, t = threadIdx.x;   // 32 threads
  float s = 0.f;
  if (t < 8) { float g = gw[bh * 8 + t]; s = expf(-g * g); }
  float tot = wave_sum(s);
  if (t < 8) al[bh * 8 + t] = s / fmaxf(tot, 1e-8f);
}

__global__ void k_transport(const float* __restrict__ Pi, const float* __restrict__ G,
                            float* __restrict__ Tb) {
  __shared__ float Gs[256];
  int bhm = blockIdx.x >> 1, hh = blockIdx.x & 1;
  int m = bhm & 7, t = threadIdx.x;
  Gs[t] = G[m * 256 + t];
  __syncthreads();
  int n = hh * 256 + t;
  const float* pr = Pi + ((size_t)bhm * Nn + n) * Kk;
  float row[16];
  for (int k = 0; k < 16; ++k) row[k] = pr[k];
  float* tr = Tb + ((size_t)bhm * Nn + n) * Kk;
  for (int l = 0; l < 16; ++l) {
    float s = 0.f;
    for (int k = 0; k < 16; ++k) s += row[k] * Gs[k * 16 + l];
    tr[l] = s;
  }
}

// Qmix[n,s] = sum_m alpha_m * T_m[n,:] . Pi_m[s,:] ; 8 waves tile 8 s-tiles,
// alpha-scaled A tile staged in LDS per m, accumulated across m in one v8f
__global__ void k_qmix(const float* __restrict__ Tb, const float* __restrict__ Pi,
                       const float* __restrict__ al, float* __restrict__ Qm) {
  __shared__ float at[16 * 16];
  int idx = blockIdx.x;                 // 64*32*4
  int bh = idx >> 7;
  int rem = idx & 127;
  int ntile = rem >> 2, schunk = rem & 3;
  int wid = threadIdx.x >> 5, lane = threadIdx.x & 31;
  int hf = lane >> 4, l16 = lane & 15;
  int n0 = ntile * 16;
  int s0 = (schunk * 8 + wid) * 16;
  v8f acc = {};
  for (int m = 0; m < 8; ++m) {
    size_t bhm = (size_t)bh * 8 + m;
    float a_m = al[bhm];
    __syncthreads();
    {
      int r = threadIdx.x >> 4, c = threadIdx.x & 15;
      at[r * 16 + c] = a_m * Tb[(bhm * Nn + n0 + r) * Kk + c];
    }
    __syncthreads();
    const float* pb = Pi + bhm * Nn * Kk;
    for (int kk = 0; kk < 4; ++kk) {
      int kl = kk * 4 + 2 * hf;
      v2f a, b;
      a.x = at[l16 * 16 + kl];
      a.y = at[l16 * 16 + kl + 1];
      b.x = pb[(size_t)(s0 + l16) * Kk + kl];       // B[l][s] = Pi[s][l]
      b.y = pb[(size_t)(s0 + l16) * Kk + kl + 1];
      acc = wmma4(a, b, acc);
    }
  }
  float* qb = Qm + (size_t)bh * Nn * Nn;
  for (int g = 0; g < 8; ++g)
    qb[(size_t)(n0 + g + 8 * hf) * Nn + s0 + l16] = acc[g];
}

// clip, row-normalize in place (Q half of d_out), bias_log = log(Q)
__global__ void k_qnorm(float* __restrict__ out) {
  __shared__ float red[8];
  size_t row = blockIdx.x;              // 0..32767
  float* q = out + (size_t)OUT_Q + row * 512;
  float* bl = out + row * 512;
  int t = threadIdx.x;
  float v0 = fmaxf(q[t], 1e-8f);
  float v1 = fmaxf(q[t + 256], 1e-8f);
  float s = block_sum(v0 + v1, red);
  float inv = 1.f / s;
  float q0 = v0 * inv, q1 = v1 * inv;
  q[t] = q0;
  q[t + 256] = q1;
  bl[t] = logf(q0);
  bl[t + 256] = logf(q1);
}

// ---------------------------------------------------------------------------
// Stage F: slot-usage regularizer + total_reg
// ---------------------------------------------------------------------------
__global__ void k_zero8(float* __restrict__ p) {
  if (threadIdx.x < 8) p[threadIdx.x] = 0.f;
}

__global__ void k_usage(const float* __restrict__ Z, const float* __restrict__ Wp,
                        const float* __restrict__ bp, float* __restrict__ usage) {
  int row = blockIdx.x * 256 + threadIdx.x;      // 4096 rows
  const float* zr = Z + (size_t)row * 256;
  float acc[8];
  for (int m = 0; m < 8; ++m) acc[m] = bp[m];
  for (int d = 0; d < 256; ++d) {
    float zv = zr[d];
    for (int m = 0; m < 8; ++m) acc[m] += zv * Wp[d * 8 + m];
  }
  float mx = acc[0];
  for (int m = 1; m < 8; ++m) mx = fmaxf(mx, acc[m]);
  float s = 0.f;
  for (int m = 0; m < 8; ++m) { acc[m] = expf(acc[m] - mx); s += acc[m]; }
  float inv = 1.f / s;
  for (int m = 0; m < 8; ++m) atomicAdd(&usage[m], acc[m] * inv);
}

__global__ void k_total(const float* __restrict__ usage, const float* __restrict__ greg,
                        float* __restrict__ out) {
  if (threadIdx.x == 0) {
    float kl = 0.f;
    for (int m = 0; m < 8; ++m) {
      float u = usage[m] * (1.f / 4096.f);
      kl += u * logf(fmaxf(u, 1e-8f) * 8.f);
    }
    out[OUT_REG] = greg[0] + 0.01f * kl;
  }
}

// ---------------------------------------------------------------------------
extern "C" void kernel_launch(void* const* d_in, const int* in_sizes, int n_in,
                              void* d_out, int out_size, void* d_ws, size_t ws_size,
                              hipStream_t stream) {
  (void)in_sizes; (void)n_in; (void)out_size; (void)ws_size;
  const float* x_desc = (const float*)d_in[0];
  const float* x_name = (const float*)d_in[1];
  const float* W1 = (const float*)d_in[2];
  const float* b1 = (const float*)d_in[3];
  const float* W2 = (const float*)d_in[4];
  const float* b2 = (const float*)d_in[5];
  const float* Wp = (const float*)d_in[6];
  const float* bp = (const float*)d_in[7];
  const float* Up = (const float*)d_in[8];
  float* out = (float*)d_out;
  float* ws = (float*)d_ws;

  // Stage A: fusion MLP
  k_concat<<<8192, 256, 0, stream>>>(x_desc, x_name, ws + OF_XCAT);
  k_gemm<<<512, 256, 0, stream>>>(ws + OF_XCAT, W1, b1, ws + OF_H1, 512, 256, 1);
  k_gemm<<<512, 256, 0, stream>>>(ws + OF_H1, W2, b2, ws + OF_Z, 256, 256, 0);

  // Stage B: affinity -> DP
  k_znorm<<<4096, 256, 0, stream>>>(ws + OF_Z, ws + OF_ZN);
  k_affinity<<<2048, 256, 0, stream>>>(ws + OF_ZN, ws + OF_P);
  k_sym_dp<<<2048, 256, 0, stream>>>(ws + OF_P, ws + OF_DP);

  // Stage C: slot graphs (H1 region now free)
  k_slots<<<1, 256, 0, stream>>>(Up, ws + OF_G, ws + OF_DG, ws + OF_DY2,
                                 ws + OF_GREG);

  // Stage D: GW loop (P region now free -> Pi/C/T; XCAT region -> masses)
  k_fill<<<16384, 256, 0, stream>>>(ws + OF_PI, A_W * B_W, BHM_ * Nn * Kk);
  for (int it = 0; it < 3; ++it) {
    k_mass<<<BHM_, 256, 0, stream>>>(ws + OF_PI, ws + OF_MJ, ws + OF_ML);
    k_term2<<<32, 256, 0, stream>>>(ws + OF_DY2, ws + OF_ML, ws + OF_T2);
    k_term1<<<2048, 256, 0, stream>>>(ws + OF_DP, ws + OF_MJ, ws + OF_T1);
    k_piDG<<<2 * BHM_, 256, 0, stream>>>(ws + OF_PI, ws + OF_DG, ws + OF_T);
    k_cross<<<2048, 256, 0, stream>>>(ws + OF_DP, ws + OF_T, ws + OF_T1,
                                      ws + OF_T2, ws + OF_C);
    k_sinkhorn<<<BHM_, 256, 0, stream>>>(ws + OF_C, ws + OF_PI, ws + OF_GW);
  }

  // Stage E: alpha, T, Qmix, normalize
  k_alpha<<<BH_, 32, 0, stream>>>(ws + OF_GW, ws + OF_AL);
  k_transport<<<2 * BHM_, 256, 0, stream>>>(ws + OF_PI, ws + OF_G, ws + OF_T);
  k_qmix<<<8192, 256, 0, stream>>>(ws + OF_T, ws + OF_PI, ws + OF_AL, out + OUT_Q);
  k_qnorm<<<32768, 256, 0, stream>>>(out);

  // Stage F: usage regularizer + total
  k_zero8<<<1, 32, 0, stream>>>(ws + OF_US);
  k_usage<<<16, 256, 0, stream>>>(ws + OF_Z, Wp, bp, ws + OF_US);
  k_total<<<1, 32, 0, stream>>>(ws + OF_US, ws + OF_GREG, out);
}